// Word2ManSkipGramOneHot_60833916781031
// MI455X (gfx1250) — compile-verified
//
#include <hip/hip_runtime.h>
#include <hip/hip_bf16.h>
#include <math.h>

#define B_SZ 1024
#define V_SZ 50257
#define E_SZ 128

typedef __attribute__((ext_vector_type(2))) float v2f;
typedef __attribute__((ext_vector_type(8))) float v8f;

// ---------------------------------------------------------------------------
// Kernel 1: per row b, find the one-hot index in xs[b,:], gather EMBEDM[idx,:],
// then t[b,:] = emb @ metric  (exactly equal to (xs @ EMBEDM) @ metric).
// ---------------------------------------------------------------------------
__global__ void k1_gather_project(const float* __restrict__ xs,
                                  const float* __restrict__ metric,
                                  const float* __restrict__ EMBEDM,
                                  float* __restrict__ t) {
    __shared__ int   s_idx;
    __shared__ float s_emb[E_SZ];
    const int b   = blockIdx.x;
    const int tid = threadIdx.x;
    if (tid == 0) s_idx = 0;
    __syncthreads();
    const float* row = xs + (size_t)b * V_SZ;
    for (int v = tid; v < V_SZ; v += blockDim.x) {
        if (row[v] > 0.5f) s_idx = v;   // exactly one lane ever hits
    }
    __syncthreads();
    const int idx = s_idx;
    if (tid < E_SZ) s_emb[tid] = EMBEDM[(size_t)idx * E_SZ + tid];
    __syncthreads();
    if (tid < E_SZ) {
        float acc = 0.f;
        #pragma unroll 8
        for (int k = 0; k < E_SZ; ++k)
            acc = fmaf(s_emb[k], metric[k * E_SZ + tid], acc);
        t[(size_t)b * E_SZ + tid] = acc;
    }
}

// ---------------------------------------------------------------------------
// Kernel 2: scores = t @ NEGEMBEDM^T via V_WMMA_F32_16X16X4_F32.
// One wave owns a 16x16 output tile; K=128 as 32 chained k=4 WMMAs.
// A operand: lane L -> t[row_base + L%16][k0 + 2*(L/16) .. +1]   (float2)
// B operand: lane L -> NEG[col_base + L%16][k0 + 2*(L/16) .. +1] (float2)
// Scores are written straight into d_out (consumed in-place by kernel 3).
// ---------------------------------------------------------------------------
__global__ void k2_scores_wmma(const float* __restrict__ t,
                               const float* __restrict__ NEG,
                               float* __restrict__ scores) {
    const int lane     = threadIdx.x & 31;
    const int wave     = threadIdx.x >> 5;
    const int row_base = blockIdx.y * 16;
    const int col_base = blockIdx.x * 128 + wave * 16;
    const int half     = lane >> 4;     // 0 or 1
    const int l16      = lane & 15;

    const int col  = col_base + l16;
    const int colc = (col < V_SZ) ? col : (V_SZ - 1);   // clamp OOB loads

    const float* aptr = t   + (size_t)(row_base + l16) * E_SZ + 2 * half;
    const float* bptr = NEG + (size_t)colc            * E_SZ + 2 * half;

    v8f c = {};
    #pragma unroll 8
    for (int s = 0; s < 32; ++s) {
        v2f a = *(const v2f*)(aptr + 4 * s);
        v2f b = *(const v2f*)(bptr + 4 * s);
        // (neg_a, A, neg_b, B, c_mod, C, reuse_a, reuse_b)
        c = __builtin_amdgcn_wmma_f32_16x16x4_f32(false, a, false, b,
                                                  (short)0, c, false, false);
    }

    if (col < V_SZ) {                    // predicate only the stores
        #pragma unroll
        for (int r = 0; r < 8; ++r) {
            const int m = r + 8 * half;  // D VGPR r holds row m, col l16
            scores[(size_t)(row_base + m) * V_SZ + col] = c[r];
        }
    }
}

// ---------------------------------------------------------------------------
// Kernel 3: in-place row log-softmax on d_out.
// Pass 1: online (max, sum-exp) per thread, tree-merge in LDS.
// Pass 2: subtract lse.
// ---------------------------------------------------------------------------
__global__ void k3_log_softmax(float* __restrict__ out) {
    const int b   = blockIdx.x;
    const int tid = threadIdx.x;
    float* row = out + (size_t)b * V_SZ;

    float m = -INFINITY, s = 0.f;
    for (int v = tid; v < V_SZ; v += blockDim.x) {
        const float x = row[v];
        if (x > m) { s = s * __expf(m - x) + 1.f; m = x; }
        else       { s += __expf(x - m); }
    }

    __shared__ float sm[256], ss[256];
    sm[tid] = m; ss[tid] = s;
    __syncthreads();
    for (int off = 128; off > 0; off >>= 1) {
        if (tid < off) {
            const float m2 = sm[tid + off], s2 = ss[tid + off];
            const float mm = fmaxf(sm[tid], m2);
            ss[tid] = ss[tid] * __expf(sm[tid] - mm) + s2 * __expf(m2 - mm);
            sm[tid] = mm;
        }
        __syncthreads();
    }
    const float lse = sm[0] + __logf(ss[0]);

    for (int v = tid; v < V_SZ; v += blockDim.x)
        row[v] = row[v] - lse;
}

// ---------------------------------------------------------------------------
extern "C" void kernel_launch(void* const* d_in, const int* in_sizes, int n_in,
                              void* d_out, int out_size, void* d_ws, size_t ws_size,
                              hipStream_t stream) {
    (void)in_sizes; (void)n_in; (void)out_size; (void)ws_size;
    const float* xs     = (const float*)d_in[0];
    const float* metric = (const float*)d_in[1];
    const float* EMBEDM = (const float*)d_in[2];
    const float* NEG    = (const float*)d_in[3];
    float*       out    = (float*)d_out;
    float*       t      = (float*)d_ws;   // 1024*128 f32 = 512 KB

    k1_gather_project<<<B_SZ, 256, 0, stream>>>(xs, metric, EMBEDM, t);

    dim3 g2((V_SZ + 127) / 128, B_SZ / 16);   // 393 x 64 workgroups, 8 waves each
    k2_scores_wmma<<<g2, 256, 0, stream>>>(t, NEG, out);

    k3_log_softmax<<<B_SZ, 256, 0, stream>>>(out);
}